// LightGCN_44298292691344
// MI455X (gfx1250) — compile-verified
//
#include <hip/hip_runtime.h>
#include <hip/hip_bf16.h>

// ---------------------------------------------------------------------------
// LightGCN propagation on MI455X (gfx1250, wave32).
//
// out = (emb + h1 + h2 + h3) / 4,  h_{k+1}[dst] += val * h_k[src]
//
// Memory/atomic-bound SpMM; the ~230MB working set is mostly L2-resident
// (192MB). fp32 throughout. Per-edge scaling (diag(vals) x X_tile) runs on
// the matrix unit via V_WMMA_F32_16X16X4_F32, one wave per 16-edge group.
// All global gathers/atomics use 32-bit byte offsets so the backend emits
// SADDR-form addressing (SGPR base + u32 VGPR offset) instead of 64-bit
// VALU address chains.
// ---------------------------------------------------------------------------

typedef float v2f __attribute__((ext_vector_type(2)));
typedef float v8f __attribute__((ext_vector_type(8)));

#define DDIM 64

__device__ __forceinline__ float ld_off(const float* p, unsigned byte_off) {
    return *(const float*)((const char*)p + (size_t)byte_off);
}
__device__ __forceinline__ void atomic_add_off(float* p, unsigned byte_off, float v) {
    __hip_atomic_fetch_add((float*)((char*)p + (size_t)byte_off), v,
                           __ATOMIC_RELAXED, __HIP_MEMORY_SCOPE_AGENT);
}

// One wave handles 16 edges: D[16x64] = diag(vals) * X_gather[16x64],
// then scatter-atomic rows of D into out[dst[e]*64 + :].
__global__ __launch_bounds__(256) void spmm_scatter_wmma(
    const float* __restrict__ x,      // [n_users, 64]
    const float* __restrict__ vals,   // [n_edges]
    const int*   __restrict__ src,    // [n_edges]
    const int*   __restrict__ dst,    // [n_edges]
    float* __restrict__ out,          // [n_users, 64], pre-zeroed
    int n_groups)                     // n_edges / 16
{
    const int lane  = threadIdx.x & 31;
    const int m     = lane & 15;      // edge-row this lane owns (both halves mirror)
    const int half  = lane >> 4;      // 0: lanes 0-15, 1: lanes 16-31
    const int waves_per_block = blockDim.x >> 5;
    const int wave_id     = blockIdx.x * waves_per_block + (threadIdx.x >> 5);
    const int total_waves = gridDim.x * waves_per_block;

    for (int g = wave_id; g < n_groups; g += total_waves) {
        const int base   = g << 4;
        const int e_mine = base + m;            // lanes 16-31 duplicate 0-15 (L0 hit)

        const float vval = vals[e_mine];
        const int   vsrc = src[e_mine];
        const int   vdst = dst[e_mine];

        // Prefetch next group's edge triples (global_prefetch_b8).
        {
            const long enext = (long)e_mine + ((long)total_waves << 4);
            __builtin_prefetch(&vals[enext], 0, 3);
            __builtin_prefetch(&src[enext], 0, 3);
            __builtin_prefetch(&dst[enext], 0, 3);
        }

        // B operand source-row byte offsets: for K-slice s, VGPR v, this lane
        // holds row k = 4s + v + 2*half of the gathered 16x64 tile.
        // Row byte offset = src_row * 64 floats * 4B = src_row << 8 (< 2^31).
        unsigned brow[4][2];
#pragma unroll
        for (int s = 0; s < 4; ++s) {
#pragma unroll
            for (int v = 0; v < 2; ++v)
                brow[s][v] = ((unsigned)__shfl(vsrc, 4 * s + v + 2 * half, 32)) << 8;
        }

        // A operand: K-slice s of diag(vals). Lane holds A[m][K], K = v + 2*half
        // within the slice; nonzero iff global column 4s+K == m.
        v2f a[4];
#pragma unroll
        for (int s = 0; s < 4; ++s) {
            a[s].x = (m == 4 * s + 0 + 2 * half) ? vval : 0.0f;
            a[s].y = (m == 4 * s + 1 + 2 * half) ? vval : 0.0f;
        }

        // Scatter-target row byte offsets: D VGPR r holds edge row (r + 8*half).
        unsigned drow[8];
#pragma unroll
        for (int r = 0; r < 8; ++r)
            drow[r] = ((unsigned)__shfl(vdst, r + 8 * half, 32)) << 8;

        // 4 column-blocks of 16 dims; each accumulates 4 K-slices.
#pragma unroll
        for (int j = 0; j < 4; ++j) {
            const unsigned colb = (unsigned)(j * 16 + m) * 4u;  // byte offset in row
            v8f c = {};
#pragma unroll
            for (int s = 0; s < 4; ++s) {
                v2f b;
                b.x = ld_off(x, brow[s][0] + colb);   // 64B segments per half-wave
                b.y = ld_off(x, brow[s][1] + colb);
                c = __builtin_amdgcn_wmma_f32_16x16x4_f32(
                        /*neg_a=*/false, a[s], /*neg_b=*/false, b,
                        /*c_mod=*/(short)0, c, /*reuse_a=*/false, /*reuse_b=*/false);
            }
#pragma unroll
            for (int r = 0; r < 8; ++r)
                atomic_add_off(out, drow[r] + colb, c[r]);
        }
    }
}

// Scalar tail for edge counts not divisible by 16 (not hit for E = 3.2M).
__global__ void spmm_scatter_tail(
    const float* __restrict__ x, const float* __restrict__ vals,
    const int* __restrict__ src, const int* __restrict__ dst,
    float* __restrict__ out, int e0, int e1)
{
    int t  = blockIdx.x * blockDim.x + threadIdx.x;
    int nd = (e1 - e0) * DDIM;
    if (t < nd) {
        int e = e0 + t / DDIM;
        int d = t % DDIM;
        float v = vals[e] * ld_off(x, (((unsigned)src[e]) << 8) + (unsigned)d * 4u);
        atomic_add_off(out, (((unsigned)dst[e]) << 8) + (unsigned)d * 4u, v);
    }
}

__global__ __launch_bounds__(256) void ew_add2(
    const float4* __restrict__ a, const float4* __restrict__ b,
    float4* __restrict__ o, int n4)
{
    for (int i = blockIdx.x * blockDim.x + threadIdx.x; i < n4;
         i += gridDim.x * blockDim.x) {
        float4 av = a[i], bv = b[i];
        o[i] = make_float4(av.x + bv.x, av.y + bv.y, av.z + bv.z, av.w + bv.w);
    }
}

__global__ __launch_bounds__(256) void ew_addinto(
    float4* __restrict__ o, const float4* __restrict__ h, int n4)
{
    for (int i = blockIdx.x * blockDim.x + threadIdx.x; i < n4;
         i += gridDim.x * blockDim.x) {
        float4 ov = o[i], hv = h[i];
        o[i] = make_float4(ov.x + hv.x, ov.y + hv.y, ov.z + hv.z, ov.w + hv.w);
    }
}

__global__ __launch_bounds__(256) void ew_finalize(
    float4* __restrict__ o, const float4* __restrict__ h, int n4, float scale)
{
    for (int i = blockIdx.x * blockDim.x + threadIdx.x; i < n4;
         i += gridDim.x * blockDim.x) {
        float4 ov = o[i], hv = h[i];
        o[i] = make_float4((ov.x + hv.x) * scale, (ov.y + hv.y) * scale,
                           (ov.z + hv.z) * scale, (ov.w + hv.w) * scale);
    }
}

extern "C" void kernel_launch(void* const* d_in, const int* in_sizes, int n_in,
                              void* d_out, int out_size, void* d_ws, size_t ws_size,
                              hipStream_t stream) {
    (void)n_in; (void)out_size; (void)ws_size;

    const float* emb  = (const float*)d_in[0];
    const float* vals = (const float*)d_in[1];
    const int*   src  = (const int*)d_in[2];
    const int*   dst  = (const int*)d_in[3];
    float*       out  = (float*)d_out;

    const int n_emb   = in_sizes[0];          // n_users * 64
    const int n_edges = in_sizes[1];
    const int n4      = n_emb / 4;

    float* hA = (float*)d_ws;                 // two [n_users,64] ping-pong buffers
    float* hB = hA + n_emb;
    const size_t emb_bytes = (size_t)n_emb * sizeof(float);

    const int n_groups  = n_edges >> 4;
    const int tail0     = n_groups << 4;
    const int tail_elts = (n_edges - tail0) * DDIM;

    dim3 blk(256);
    dim3 grdS(2048);   // 16384 waves over 200k groups
    dim3 grdE(2048);
    dim3 grdT(tail_elts > 0 ? (tail_elts + 255) / 256 : 1);

    // ---- layer 1: h1 = G * emb ; out = emb + h1
    hipMemsetAsync(hA, 0, emb_bytes, stream);
    spmm_scatter_wmma<<<grdS, blk, 0, stream>>>(emb, vals, src, dst, hA, n_groups);
    if (tail_elts > 0)
        spmm_scatter_tail<<<grdT, blk, 0, stream>>>(emb, vals, src, dst, hA, tail0, n_edges);
    ew_add2<<<grdE, blk, 0, stream>>>((const float4*)emb, (const float4*)hA,
                                      (float4*)out, n4);

    // ---- layer 2: h2 = G * h1 ; out += h2
    hipMemsetAsync(hB, 0, emb_bytes, stream);
    spmm_scatter_wmma<<<grdS, blk, 0, stream>>>(hA, vals, src, dst, hB, n_groups);
    if (tail_elts > 0)
        spmm_scatter_tail<<<grdT, blk, 0, stream>>>(hA, vals, src, dst, hB, tail0, n_edges);
    ew_addinto<<<grdE, blk, 0, stream>>>((float4*)out, (const float4*)hB, n4);

    // ---- layer 3: h3 = G * h2 ; out = (out + h3) * 0.25
    hipMemsetAsync(hA, 0, emb_bytes, stream);
    spmm_scatter_wmma<<<grdS, blk, 0, stream>>>(hB, vals, src, dst, hA, n_groups);
    if (tail_elts > 0)
        spmm_scatter_tail<<<grdT, blk, 0, stream>>>(hB, vals, src, dst, hA, tail0, n_edges);
    ew_finalize<<<grdE, blk, 0, stream>>>((float4*)out, (const float4*)hA, n4, 0.25f);
}